// AutoEncoder_31233002176631
// MI455X (gfx1250) — compile-verified
//
#include <hip/hip_runtime.h>

#define EPS_BN 1e-5f

typedef __attribute__((ext_vector_type(2))) float v2f;
typedef __attribute__((ext_vector_type(8))) float v8f;

// ---------- small utility kernels ----------

__global__ void fill_zero_kernel(float* __restrict__ p, long long n) {
  long long i = (long long)blockIdx.x * blockDim.x + threadIdx.x;
  if (i < n) p[i] = 0.0f;
}

__global__ void neg_copy_kernel(float* __restrict__ dst, const float* __restrict__ src, long long n) {
  long long i = (long long)blockIdx.x * blockDim.x + threadIdx.x;
  if (i < n) dst[i] = -src[i];
}

__global__ void concat_kernel(const float* __restrict__ x, const float* __restrict__ pos,
                              const float* __restrict__ nor, float* __restrict__ h0, int n) {
  int i = blockIdx.x * blockDim.x + threadIdx.x;
  if (i >= n) return;
  float* o = h0 + (size_t)i * 9;
  const size_t b = (size_t)i * 3;
  o[0] = x[b]; o[1] = x[b + 1]; o[2] = x[b + 2];
  o[3] = pos[b]; o[4] = pos[b + 1]; o[5] = pos[b + 2];
  o[6] = nor[b]; o[7] = nor[b + 1]; o[8] = nor[b + 2];
}

// ---------- graph-Laplacian weights ----------

__global__ void deg_kernel(const int* __restrict__ src, const int* __restrict__ dst,
                           float* __restrict__ deg, int E) {
  int e = blockIdx.x * blockDim.x + threadIdx.x;
  if (e < E) {
    int s = src[e];
    if (s != dst[e]) atomicAdd(&deg[s], 1.0f);
  }
}

__global__ void dinv_kernel(float* __restrict__ deg, int n) {
  int i = blockIdx.x * blockDim.x + threadIdx.x;
  if (i < n) {
    float d = deg[i];
    deg[i] = (d > 0.0f) ? rsqrtf(d) : 0.0f;   // in-place deg -> dinv
  }
}

__global__ void edgew_kernel(const int* __restrict__ src, const int* __restrict__ dst,
                             const float* __restrict__ dinv, float* __restrict__ w, int E) {
  int e = blockIdx.x * blockDim.x + threadIdx.x;
  if (e < E) {
    int s = src[e], d = dst[e];
    w[e] = (s != d) ? (-dinv[s] * dinv[d]) : 0.0f;
  }
}

// ---------- sparse propagation (scatter-add) ----------

// C = 9 variant: one thread per edge
__global__ void prop9_kernel(const float* __restrict__ in, float* __restrict__ out,
                             const int* __restrict__ src, const int* __restrict__ dst,
                             const float* __restrict__ w, float scale, int E) {
  int e = blockIdx.x * blockDim.x + threadIdx.x;
  if (e >= E) return;
  float ww = w[e];
  if (ww == 0.0f) return;
  ww *= scale;
  const float* ip = in + (size_t)src[e] * 9;
  float* op = out + (size_t)dst[e] * 9;
#pragma unroll
  for (int j = 0; j < 9; ++j) atomicAdd(&op[j], ww * ip[j]);
}

// C = 256 variant: blockIdx.y selects a 4-channel group, one thread per edge
__global__ void prop_c4_kernel(const float* __restrict__ in, float* __restrict__ out,
                               const int* __restrict__ src, const int* __restrict__ dst,
                               const float* __restrict__ w, float scale, int E, int C) {
  int e = blockIdx.x * blockDim.x + threadIdx.x;
  if (e >= E) return;
  float ww = w[e];
  if (ww == 0.0f) return;
  ww *= scale;
  const int cg = blockIdx.y << 2;
  const float4 v = *(const float4*)(in + (size_t)src[e] * C + cg);
  float* o = out + (size_t)dst[e] * C + cg;
  atomicAdd(o + 0, ww * v.x);
  atomicAdd(o + 1, ww * v.y);
  atomicAdd(o + 2, ww * v.z);
  atomicAdd(o + 3, ww * v.w);
}

// ---------- layer-1 dense (K=9, plain VALU) ----------

__global__ __launch_bounds__(256)
void cheb_gemm_l1_kernel(const float* __restrict__ h0, const float* __restrict__ t1,
                         const float* __restrict__ t2, const float* __restrict__ W,
                         const float* __restrict__ bias, const float* __restrict__ slope,
                         float* __restrict__ out, float* __restrict__ stats) {
  __shared__ float lh[3][32 * 9];
  const int c = threadIdx.x;          // output channel 0..255
  const int r0 = blockIdx.x * 32;     // 32 nodes per block (N % 32 == 0)
  for (int idx = c; idx < 32 * 9; idx += 256) {
    lh[0][idx] = h0[(size_t)r0 * 9 + idx];
    lh[1][idx] = t1[(size_t)r0 * 9 + idx];
    lh[2][idx] = t2[(size_t)r0 * 9 + idx];
  }
  __syncthreads();
  float Wr[27];
#pragma unroll
  for (int m = 0; m < 3; ++m)
#pragma unroll
    for (int k = 0; k < 9; ++k) Wr[m * 9 + k] = W[(size_t)(m * 9 + k) * 256 + c];
  const float b = bias[c];
  const float al = slope[0];
  float s = 0.0f, q = 0.0f;
  for (int r = 0; r < 32; ++r) {
    float v = b;
#pragma unroll
    for (int m = 0; m < 3; ++m)
#pragma unroll
      for (int k = 0; k < 9; ++k) v = fmaf(lh[m][r * 9 + k], Wr[m * 9 + k], v);
    v = (v >= 0.0f) ? v : al * v;
    out[(size_t)(r0 + r) * 256 + c] = v;
    s += v;
    q += v * v;
  }
  atomicAdd(&stats[c], s);
  atomicAdd(&stats[256 + c], q);
}

// ---------- WMMA fp32 GEMM: out = h@W0 + tx1@W1 + tx2@W2 + b, PReLU, BN stats ----------
// Each wave computes a 16x32 output tile (two 16x16 WMMA accumulators sharing A fragments).
// The staged weight panels use a k-pair-interleaved LDS layout [m][k/2][n][2] so that each
// B fragment (B[kk][n], B[kk+1][n]) is one aligned 8-byte ds_load_b64 into a VGPR pair.

__device__ __forceinline__ float swapx16(float v) {
  // ds_swizzle group-of-32: xor=0x10, or=0, and=0x1f  -> exchange lane l <-> l^16
  return __int_as_float(__builtin_amdgcn_ds_swizzle(__float_as_int(v), 0x401F));
}

template <int COUT>
__global__ __launch_bounds__(256)
void cheb_gemm_wmma_kernel(const float* __restrict__ A0, const float* __restrict__ A1,
                           const float* __restrict__ A2, const float* __restrict__ W,
                           const float* __restrict__ bias, const float* __restrict__ slope,
                           float* __restrict__ out, float* __restrict__ stats, int nRowTiles) {
  constexpr int K = 256;
  constexpr int MSTRIDE = (K / 2) * 64;  // floats per staged weight matrix panel
  __shared__ float ldsB[3 * MSTRIDE];    // 96 KB total

  const int tid = threadIdx.x;
  const int n0 = blockIdx.x * 32;

  // Stage B pair-interleaved: ldsB[m*MSTRIDE + k2*64 + n*2 + j] = W[m][2*k2+j][n0+n]
  for (int idx = tid; idx < 3 * MSTRIDE; idx += 256) {
    int m = idx / MSTRIDE;
    int rem = idx - m * MSTRIDE;
    int k2 = rem >> 6;
    int jn = rem & 63;
    int n = jn >> 1;
    int k = (k2 << 1) | (jn & 1);
    ldsB[idx] = W[((size_t)m * K + k) * COUT + n0 + n];
  }
  __syncthreads();

  const int wave = tid >> 5;
  const int lane = tid & 31;
  const int rt = blockIdx.y * 8 + wave;
  if (rt >= nRowTiles) return;

  const int row0 = rt * 16;
  const int m16 = lane & 15;            // A row / B col within tile
  const int kb = (lane >> 4) << 1;      // K sub-offset: 0 (lanes 0-15) or 2 (lanes 16-31)

  const float* a0p = A0 + (size_t)(row0 + m16) * K;
  const float* a1p = A1 + (size_t)(row0 + m16) * K;
  const float* a2p = A2 + (size_t)(row0 + m16) * K;
  // Per-lane B fragment bases: matrix panel + k-pair for this half-wave + column.
  const int laneB = (kb >> 1) * 64 + m16 * 2;
  const float* BL0 = &ldsB[0 * MSTRIDE + laneB];   // cols n0+0..15
  const float* BL1 = &ldsB[1 * MSTRIDE + laneB];
  const float* BL2 = &ldsB[2 * MSTRIDE + laneB];
  const float* BH0 = BL0 + 32;                     // cols n0+16..31
  const float* BH1 = BL1 + 32;
  const float* BH2 = BL2 + 32;

  v8f accL = {};
  v8f accH = {};
  for (int k0 = 0; k0 < K; k0 += 4) {
    const int kk = k0 + kb;
    const int o = k0 * 32;  // float offset into pair-interleaved panels
    v2f a0 = *(const v2f*)(a0p + kk);
    v2f a1 = *(const v2f*)(a1p + kk);
    v2f a2 = *(const v2f*)(a2p + kk);
    v2f bL0 = *(const v2f*)(BL0 + o);
    v2f bH0 = *(const v2f*)(BH0 + o);
    v2f bL1 = *(const v2f*)(BL1 + o);
    v2f bH1 = *(const v2f*)(BH1 + o);
    v2f bL2 = *(const v2f*)(BL2 + o);
    v2f bH2 = *(const v2f*)(BH2 + o);
    accL = __builtin_amdgcn_wmma_f32_16x16x4_f32(false, a0, false, bL0, (short)0, accL, false, false);
    accH = __builtin_amdgcn_wmma_f32_16x16x4_f32(false, a0, false, bH0, (short)0, accH, false, false);
    accL = __builtin_amdgcn_wmma_f32_16x16x4_f32(false, a1, false, bL1, (short)0, accL, false, false);
    accH = __builtin_amdgcn_wmma_f32_16x16x4_f32(false, a1, false, bH1, (short)0, accH, false, false);
    accL = __builtin_amdgcn_wmma_f32_16x16x4_f32(false, a2, false, bL2, (short)0, accL, false, false);
    accH = __builtin_amdgcn_wmma_f32_16x16x4_f32(false, a2, false, bH2, (short)0, accH, false, false);
  }

  // Epilogue: bias, PReLU, store, BN partial sums (both 16-col sub-tiles)
  const float al = slope[0];
  const int rowBase = row0 + ((lane >> 4) << 3);  // lanes 16-31 hold M = r+8
  const float bnL = bias[n0 + m16];
  const float bnH = bias[n0 + 16 + m16];
  float sL = 0.0f, qL = 0.0f, sH = 0.0f, qH = 0.0f;
#pragma unroll
  for (int r = 0; r < 8; ++r) {
    float vL = accL[r] + bnL;
    vL = (vL >= 0.0f) ? vL : al * vL;
    out[(size_t)(rowBase + r) * COUT + n0 + m16] = vL;
    sL += vL;
    qL += vL * vL;
    float vH = accH[r] + bnH;
    vH = (vH >= 0.0f) ? vH : al * vH;
    out[(size_t)(rowBase + r) * COUT + n0 + 16 + m16] = vH;
    sH += vH;
    qH += vH * vH;
  }
  sL += swapx16(sL);
  qL += swapx16(qL);
  sH += swapx16(sH);
  qH += swapx16(qH);
  if (lane < 16) {
    atomicAdd(&stats[n0 + m16], sL);
    atomicAdd(&stats[256 + n0 + m16], qL);
    atomicAdd(&stats[n0 + 16 + m16], sH);
    atomicAdd(&stats[256 + n0 + 16 + m16], qH);
  }
}

// ---------- BatchNorm finalize + apply ----------

__global__ void bn_finalize_kernel(const float* __restrict__ stats, const float* __restrict__ g,
                                   const float* __restrict__ be, float* __restrict__ ss,
                                   int C, float invN) {
  int c = blockIdx.x * blockDim.x + threadIdx.x;
  if (c >= C) return;
  float mean = stats[c] * invN;
  float var = stats[256 + c] * invN - mean * mean;
  float inv = rsqrtf(var + EPS_BN);
  float sc = inv * g[c];
  ss[c] = sc;
  ss[256 + c] = be[c] - mean * sc;
}

__global__ void bn_apply_kernel(const float* __restrict__ in, float* __restrict__ out,
                                const float* __restrict__ ss, long long n, int cmask) {
  long long i = (long long)blockIdx.x * blockDim.x + threadIdx.x;
  if (i >= n) return;
  int c = (int)(i & (long long)cmask);
  out[i] = fmaf(in[i], ss[c], ss[256 + c]);
}

// ---------- orchestration ----------

static inline dim3 gb(long long n) { return dim3((unsigned)((n + 255) / 256)); }

extern "C" void kernel_launch(void* const* d_in, const int* in_sizes, int n_in,
                              void* d_out, int out_size, void* d_ws, size_t ws_size,
                              hipStream_t stream) {
  const float* x = (const float*)d_in[0];
  const float* pos = (const float*)d_in[1];
  const float* nor = (const float*)d_in[2];
  const int* ei = (const int*)d_in[3];
  const int N = in_sizes[0] / 3;
  const int E = in_sizes[3] / 2;
  const int* src = ei;
  const int* dst = ei + E;

  const float *W[4], *b[4], *a[4], *g[4], *be[4];
  for (int i = 0; i < 4; ++i) {
    W[i] = (const float*)d_in[4 + 5 * i];
    b[i] = (const float*)d_in[5 + 5 * i];
    a[i] = (const float*)d_in[6 + 5 * i];
    g[i] = (const float*)d_in[7 + 5 * i];
    be[i] = (const float*)d_in[8 + 5 * i];
  }

  float* ws = (float*)d_ws;
  const size_t NC = (size_t)N * 256;
  float* bufA = ws;                 // h (layer input, post-BN)
  float* bufB = ws + NC;            // tx1
  float* bufC = ws + 2 * NC;        // tx2
  float* bufD = ws + 3 * NC;        // pre-BN layer output
  float* h0 = ws + 4 * NC;          // [N,9] concat input (slot N*16)
  float* wE = h0 + (size_t)N * 16;  // per-edge Laplacian weight
  float* deg = wE + (size_t)E;      // degree -> dinv (in place)
  float* stats = deg + (size_t)N;   // sum[256] | sumsq[256]
  float* ss = stats + 512;          // scale[256] | shift[256]

  const dim3 B256(256);

  // Laplacian edge weights
  fill_zero_kernel<<<gb(N), B256, 0, stream>>>(deg, N);
  deg_kernel<<<gb(E), B256, 0, stream>>>(src, dst, deg, E);
  dinv_kernel<<<gb(N), B256, 0, stream>>>(deg, N);
  edgew_kernel<<<gb(E), B256, 0, stream>>>(src, dst, deg, wE, E);

  // Layer 1 (Cin = 9)
  concat_kernel<<<gb(N), B256, 0, stream>>>(x, pos, nor, h0, N);
  fill_zero_kernel<<<gb((long long)N * 9), B256, 0, stream>>>(bufB, (long long)N * 9);
  prop9_kernel<<<gb(E), B256, 0, stream>>>(h0, bufB, src, dst, wE, 1.0f, E);
  neg_copy_kernel<<<gb((long long)N * 9), B256, 0, stream>>>(bufC, h0, (long long)N * 9);
  prop9_kernel<<<gb(E), B256, 0, stream>>>(bufB, bufC, src, dst, wE, 2.0f, E);
  fill_zero_kernel<<<gb(512), B256, 0, stream>>>(stats, 512);
  cheb_gemm_l1_kernel<<<dim3(N / 32), B256, 0, stream>>>(h0, bufB, bufC, W[0], b[0], a[0],
                                                         bufD, stats);
  bn_finalize_kernel<<<dim3(1), B256, 0, stream>>>(stats, g[0], be[0], ss, 256, 1.0f / N);
  bn_apply_kernel<<<gb(NC), B256, 0, stream>>>(bufD, bufA, ss, (long long)NC, 255);

  // Layers 2..4 (Cin = 256, WMMA fp32)
  const int rowTiles = N / 16;
  const dim3 propGrid((unsigned)((E + 255) / 256), 64);
  for (int l = 1; l < 4; ++l) {
    const int Cout = (l == 3) ? 128 : 256;
    fill_zero_kernel<<<gb(NC), B256, 0, stream>>>(bufB, (long long)NC);
    prop_c4_kernel<<<propGrid, B256, 0, stream>>>(bufA, bufB, src, dst, wE, 1.0f, E, 256);
    neg_copy_kernel<<<gb(NC), B256, 0, stream>>>(bufC, bufA, (long long)NC);
    prop_c4_kernel<<<propGrid, B256, 0, stream>>>(bufB, bufC, src, dst, wE, 2.0f, E, 256);
    fill_zero_kernel<<<gb(512), B256, 0, stream>>>(stats, 512);
    const dim3 gemmGrid((unsigned)(Cout / 32), (unsigned)((rowTiles + 7) / 8));
    if (Cout == 256) {
      cheb_gemm_wmma_kernel<256><<<gemmGrid, B256, 0, stream>>>(bufA, bufB, bufC, W[l], b[l],
                                                                a[l], bufD, stats, rowTiles);
    } else {
      cheb_gemm_wmma_kernel<128><<<gemmGrid, B256, 0, stream>>>(bufA, bufB, bufC, W[l], b[l],
                                                                a[l], bufD, stats, rowTiles);
    }
    bn_finalize_kernel<<<dim3(1), B256, 0, stream>>>(stats, g[l], be[l], ss, Cout, 1.0f / N);
    float* dest = (l == 3) ? (float*)d_out : bufA;
    bn_apply_kernel<<<gb((long long)N * Cout), B256, 0, stream>>>(bufD, dest, ss,
                                                                  (long long)N * Cout, Cout - 1);
  }
}